// RetInd_61546881351990
// MI455X (gfx1250) — compile-verified
//
#include <hip/hip_runtime.h>
#include <math.h>

// ---------------------------------------------------------------------------
// RetInd: bilinear retrieval scoring on MI455X (gfx1250).
//   q = cur @ W                      (16x768) -- WMMA f32 16x16x4
//   act[b,m,k] = <all_ex[b,m,k,:], q[b,:]> + bias  then prefix-index masking
//   value[b] = <q[b], vw> + vb, repeated M times
// The einsum streams 786 MB fp32 -> bandwidth bound (~34us @ 23.3 TB/s).
// We still run the dot products through V_WMMA_F32_16X16X4_F32: 16 rows per
// wave, q broadcast across the 16 B-columns, b128 global loads for A via a
// K-permutation (B is permuted identically, so D is unchanged).
// ---------------------------------------------------------------------------

typedef float v2f __attribute__((ext_vector_type(2)));
typedef float v4f __attribute__((ext_vector_type(4)));
typedef float v8f __attribute__((ext_vector_type(8)));

#define EMB 768
#define BB  16
#define MM  8
#define KK  2000
#define ROWS_PER_B (MM * KK)      /* 16000 rows of length EMB per batch  */
#define TILES_PER_B (ROWS_PER_B / 16)   /* 1000 */
#define BLOCKS_PER_B (TILES_PER_B / 8)  /* 125: 8 waves per 256-thr block */

static __device__ __forceinline__ v8f wmma_f32(v2f a, v2f b, v8f c) {
  // 8 args: (neg_a, A, neg_b, B, c_mod, C, reuse_a, reuse_b)
  return __builtin_amdgcn_wmma_f32_16x16x4_f32(
      false, a, false, b, (short)0, c, false, false);
}

// ---------------------------------------------------------------------------
// Kernel 1: q[16x768] = cur[16x768] @ W[768x768].  One wave per 16-col tile.
// A layout (16x4 f32): lanes 0-15 hold row (lane&15), hw-k {0,1}; lanes 16-31
// hold hw-k {2,3}. We permute so hw-k {0,1,2,3} -> data-k {k0,k0+1,k0+4,k0+5}
// (lo half) / {..} so a single v4f load at (k0 + 4*hi) feeds two WMMAs.
// ---------------------------------------------------------------------------
__global__ __launch_bounds__(32)
void qgemm_kernel(const float* __restrict__ cur, const float* __restrict__ W,
                  float* __restrict__ qws) {
  const int lane = threadIdx.x & 31;
  const int hi   = lane >> 4;       // 0: lanes 0-15, 1: lanes 16-31
  const int lr   = lane & 15;       // A-row index AND B/D column index
  const int n0   = blockIdx.x * 16; // output column tile

  const float* aptr = cur + (size_t)lr * EMB + 4 * hi;
  const float* wcol = W + n0 + lr;

  v8f acc = {0.f, 0.f, 0.f, 0.f, 0.f, 0.f, 0.f, 0.f};
#pragma unroll 4
  for (int k0 = 0; k0 < EMB; k0 += 8) {
    v4f a = *(const v4f*)(aptr + k0);          // data-k k0..k0+3 (+4*hi)
    const int kb = (k0 + 4 * hi) * EMB;
    v2f a0; a0.x = a.x; a0.y = a.y;            // step A: data-k {k0, k0+1}
    v2f a1; a1.x = a.z; a1.y = a.w;            // step B: data-k {k0+2,k0+3}
    v2f b0; b0.x = wcol[kb];           b0.y = wcol[kb + EMB];
    v2f b1; b1.x = wcol[kb + 2 * EMB]; b1.y = wcol[kb + 3 * EMB];
    acc = wmma_f32(a0, b0, acc);
    acc = wmma_f32(a1, b1, acc);
  }

  // D 16x16: vgpr r -> row r (lanes 0-15) / row r+8 (lanes 16-31), col = lr.
  float* dst = qws + (size_t)(8 * hi) * EMB + n0 + lr;
#pragma unroll
  for (int r = 0; r < 8; ++r) dst[(size_t)r * EMB] = acc[r];
}

// ---------------------------------------------------------------------------
// Kernel 2: value estimates -> out[256000 .. 256127]
// ---------------------------------------------------------------------------
__global__ __launch_bounds__(128)
void value_kernel(const float* __restrict__ qws, const float* __restrict__ vw,
                  const float* __restrict__ vb, float* __restrict__ out) {
  const int t = threadIdx.x;        // 0..127  (b = t/8)
  const int b = t >> 3;
  const float* qrow = qws + (size_t)b * EMB;
  float s = 0.f;
#pragma unroll 4
  for (int e = 0; e < EMB; ++e) s += qrow[e] * vw[e];
  out[(size_t)BB * ROWS_PER_B + t] = s + vb[0];
}

// ---------------------------------------------------------------------------
// Kernel 3: dominant streaming kernel.  256 threads = 8 waves; each wave owns
// one 16-row tile (48 KB contiguous of all_ex[b]).  q[b] staged in LDS and
// broadcast into all 16 WMMA B-columns (conflict-free ds_load_b128).
// Epilogue: lanes 0/16 hold column 0 of D -> bias + prefix-index mask -> out.
// ---------------------------------------------------------------------------
__global__ __launch_bounds__(256)
void act_kernel(const float* __restrict__ ex, const float* __restrict__ qws,
                const int* __restrict__ idx, const float* __restrict__ bias,
                float* __restrict__ out) {
  __shared__ float qs[EMB];
  __shared__ int   sidx[MM];

  const int tid  = threadIdx.x;
  const int b    = blockIdx.x / BLOCKS_PER_B;
  const int tile = (blockIdx.x % BLOCKS_PER_B) * 8 + (tid >> 5);
  const int lane = tid & 31;
  const int hi   = lane >> 4;
  const int lr   = lane & 15;

  for (int i = tid; i < EMB; i += 256) qs[i] = qws[(size_t)b * EMB + i];
  if (tid < MM) sidx[tid] = idx[b * MM + tid];
  __syncthreads();

  const int r0 = tile * 16;
  const float* aptr =
      ex + ((size_t)b * ROWS_PER_B + r0 + lr) * EMB + 4 * hi;

  v8f acc = {0.f, 0.f, 0.f, 0.f, 0.f, 0.f, 0.f, 0.f};
#pragma unroll 4
  for (int k0 = 0; k0 < EMB; k0 += 8) {
    v4f a  = *(const v4f*)(aptr + k0);            // global_load_b128
    v4f qv = *(const v4f*)(&qs[k0 + 4 * hi]);     // ds_load_b128 broadcast
    v2f a0; a0.x = a.x;  a0.y = a.y;
    v2f a1; a1.x = a.z;  a1.y = a.w;
    v2f b0; b0.x = qv.x; b0.y = qv.y;
    v2f b1; b1.x = qv.z; b1.y = qv.w;
    acc = wmma_f32(a0, b0, acc);
    acc = wmma_f32(a1, b1, acc);
  }

  // All WMMAs done; EXEC divergence is now safe.
  const float bv = bias[0];
  if (lr == 0) {                       // lanes 0 (rows r0..r0+7) and 16 (+8)
#pragma unroll
    for (int rr = 0; rr < 8; ++rr) {
      const int row = r0 + rr + 8 * hi;      // row within this batch's 16000
      const int m = row / KK;
      const int k = row - m * KK;
      bool masked = false;
      for (int u = 0; u < m; ++u) masked = masked || (sidx[u] == k);
      out[(size_t)b * ROWS_PER_B + row] =
          masked ? -INFINITY : (acc[rr] + bv);
    }
  }
}

extern "C" void kernel_launch(void* const* d_in, const int* in_sizes, int n_in,
                              void* d_out, int out_size, void* d_ws, size_t ws_size,
                              hipStream_t stream) {
  const float* cur  = (const float*)d_in[0];
  // d_in[1] (example_encodings) is only used for its shape in the reference.
  const float* ex   = (const float*)d_in[2];
  const int*   idx  = (const int*)d_in[3];
  const float* W    = (const float*)d_in[4];
  const float* bias = (const float*)d_in[5];
  const float* vw   = (const float*)d_in[6];
  const float* vb   = (const float*)d_in[7];
  float* out = (float*)d_out;
  float* qws = (float*)d_ws;          // 16*768 floats = 48 KB scratch

  hipLaunchKernelGGL(qgemm_kernel, dim3(EMB / 16), dim3(32), 0, stream,
                     cur, W, qws);
  hipLaunchKernelGGL(value_kernel, dim3(1), dim3(128), 0, stream,
                     qws, vw, vb, out);
  hipLaunchKernelGGL(act_kernel, dim3(BB * BLOCKS_PER_B), dim3(256), 0, stream,
                     ex, qws, idx, bias, out);
}